// GriddingDistance_73169062855118
// MI455X (gfx1250) — compile-verified
//
#include <hip/hip_runtime.h>
#include <hip/hip_bf16.h>
#include <stdint.h>

// TDM descriptor groups (amdgpu-toolchain / clang-23 6-arg builtin form:
//   (uint32x4 g0, int32x8 g1, int32x4 g2, int32x4 g3, int32x8 g4, i32 cpol))
typedef __attribute__((ext_vector_type(4))) unsigned int tdm_g0_t;
typedef __attribute__((ext_vector_type(8))) int          tdm_g1_t;
typedef __attribute__((ext_vector_type(4))) int          tdm_g23_t;

#define R_GRID        128
#define PTS_PER_BLOCK 256
#define PTS_PER_WAVE  32

__global__ __launch_bounds__(256) void gridding_zero_kernel(float4* __restrict__ out, int n4) {
    int i = blockIdx.x * blockDim.x + threadIdx.x;
    int stride = gridDim.x * blockDim.x;
    float4 z; z.x = 0.f; z.y = 0.f; z.z = 0.f; z.w = 0.f;
    for (; i < n4; i += stride) out[i] = z;
}

__global__ __launch_bounds__(256) void gridding_scatter_kernel(const float* __restrict__ cloud,
                                                               float* __restrict__ grid,
                                                               int n_per_batch) {
    __shared__ float smem[PTS_PER_BLOCK * 3];

    // wave-uniform values kept scalar so the TDM descriptor lives in SGPRs
    const int wave = __builtin_amdgcn_readfirstlane((int)(threadIdx.x >> 5));
    const int lane = (int)(threadIdx.x & 31);

    const long long pbase = (long long)blockIdx.x * PTS_PER_BLOCK;   // first point of block
    const unsigned long long gaddr =
        (unsigned long long)(uintptr_t)cloud +
        (unsigned long long)(pbase + (long long)wave * PTS_PER_WAVE) * 12ull;

    const unsigned int lds_base =
        (unsigned int)(uintptr_t)(&smem[0]) + (unsigned int)(wave * (PTS_PER_WAVE * 12));

    // ---- Tensor Data Mover: DMA this wave's 32 points (96 dwords) into LDS ----
    tdm_g0_t g0;
    g0[0] = 1u;                                    // count=1, user descriptor
    g0[1] = lds_base;                              // lds_addr (bytes)
    g0[2] = (unsigned int)(gaddr & 0xFFFFFFFFull); // global_addr[31:0]
    g0[3] = (unsigned int)(gaddr >> 32) | (2u << 30); // global_addr[56:32] | type=2

    tdm_g1_t g1;
    g1[0] = 0x00020000;        // workgroup_mask=0, data_size=2 (4B), no barrier/pad/iterate
    g1[1] = (96 << 16);        // tensor_dim0 lo16 = 96 elements
    g1[2] = 0;                 // tensor_dim0 hi16 = 0, tensor_dim1 = 0
    g1[3] = (96 << 16);        // tile_dim0 = 96
    g1[4] = 0;                 // tile_dim1 = tile_dim2 = 0 (1-D tile)
    g1[5] = 96;                // tensor_dim0_stride = 96
    g1[6] = 0;
    g1[7] = 0;

    tdm_g23_t gz;  gz[0] = 0; gz[1] = 0; gz[2] = 0; gz[3] = 0;
    tdm_g1_t  gz8; gz8[0] = 0; gz8[1] = 0; gz8[2] = 0; gz8[3] = 0;
               gz8[4] = 0; gz8[5] = 0; gz8[6] = 0; gz8[7] = 0;

    __builtin_amdgcn_tensor_load_to_lds(g0, g1, gz, gz, gz8, 0);
    __builtin_amdgcn_s_wait_tensorcnt(0);
    asm volatile("" ::: "memory");   // TDM intrinsic has no pointer arg: force LDS reload

    // stride-3 over 32 lanes -> conflict-free LDS reads
    const int fi = (wave * PTS_PER_WAVE + lane) * 3;
    const float cx = smem[fi + 0];
    const float cy = smem[fi + 1];
    const float cz = smem[fi + 2];

    const float Rf  = (float)R_GRID;
    const float inv = 1.0f / Rf;                   // R is 2^7: exact
    const float vx = (cx * inv + 0.5f) * Rf;
    const float vy = (cy * inv + 0.5f) * Rf;
    const float vz = (cz * inv + 0.5f) * Rf;

    const float flx = floorf(vx), fly = floorf(vy), flz = floorf(vz);
    const int lx = (int)flx, ly = (int)fly, lz = (int)flz;
    const float fx = vx - flx, fy = vy - fly, fz = vz - flz;

    const float wx[2] = {1.0f - fx, fx};
    const float wy[2] = {1.0f - fy, fy};
    const float wz[2] = {1.0f - fz, fz};

    const int b = (int)(pbase / n_per_batch);      // blocks never straddle batches
    float* gb = grid + (size_t)b * (R_GRID * R_GRID * R_GRID);

#pragma unroll
    for (int i = 0; i < 2; ++i) {
#pragma unroll
        for (int j = 0; j < 2; ++j) {
#pragma unroll
            for (int k = 0; k < 2; ++k) {
                const int X = lx + i, Y = ly + j, Z = lz + k;
                if ((unsigned)X < (unsigned)R_GRID &&
                    (unsigned)Y < (unsigned)R_GRID &&
                    (unsigned)Z < (unsigned)R_GRID) {
                    const float w = wx[i] * wy[j] * wz[k];
                    const int idx = (X * R_GRID + Y) * R_GRID + Z;
                    // no-return f32 atomic -> global_atomic_add_f32, resolves in L2
                    __hip_atomic_fetch_add(gb + idx, w, __ATOMIC_RELAXED,
                                           __HIP_MEMORY_SCOPE_AGENT);
                }
            }
        }
    }
}

extern "C" void kernel_launch(void* const* d_in, const int* in_sizes, int n_in,
                              void* d_out, int out_size, void* d_ws, size_t ws_size,
                              hipStream_t stream) {
    const float* pred = (const float*)d_in[0];
    const float* gt   = (const float*)d_in[1];
    float* out = (float*)d_out;

    const int B = 16;
    const int N1 = in_sizes[0] / (3 * B);
    const int N2 = in_sizes[1] / (3 * B);
    const size_t cells_per_cloud = (size_t)B * R_GRID * R_GRID * R_GRID;

    // 1) zero both output grids (256 MB) with wide stores
    const int n4 = out_size / 4;
    gridding_zero_kernel<<<4096, 256, 0, stream>>>((float4*)out, n4);

    // 2) scatter each cloud; batch-contiguous block order keeps each 8MB grid
    //    slice resident in the 192MB L2 while its atomics land
    gridding_scatter_kernel<<<(B * N1) / PTS_PER_BLOCK, PTS_PER_BLOCK, 0, stream>>>(
        pred, out, N1);
    gridding_scatter_kernel<<<(B * N2) / PTS_PER_BLOCK, PTS_PER_BLOCK, 0, stream>>>(
        gt, out + cells_per_cloud, N2);
}